// EvalSH_58282706207075
// MI455X (gfx1250) — compile-verified
//
#include <hip/hip_runtime.h>

typedef __attribute__((ext_vector_type(2))) float v2f;
typedef __attribute__((ext_vector_type(4))) float v4f;
typedef __attribute__((ext_vector_type(8))) float v8f;

// Register-indexed extract from the 8-VGPR WMMA accumulator (diagonal pick).
// r is in [0,7]; lowered as a 7-deep cmp/cndmask chain (compares shared
// across the two channel accumulators by the compiler).
__device__ __forceinline__ float diag_pick8(const v8f a, int r) {
  float v = a[0];
  v = (r == 1) ? a[1] : v;
  v = (r == 2) ? a[2] : v;
  v = (r == 3) ? a[3] : v;
  v = (r == 4) ? a[4] : v;
  v = (r == 5) ? a[5] : v;
  v = (r == 6) ? a[6] : v;
  v = (r == 7) ? a[7] : v;
  return v;
}

// One wave32 evaluates degree-3 SH (16 coeffs, 2 channels) for a 16-point tile.
// out[p][c] = sum_k basis[p][k] * S[p][k][c] runs on the matrix pipe as
// D = basis(16x16) x B with B[k][j] = S[point j][k][c]; diag(D) is the answer.
// K=16 -> 4 chained V_WMMA_F32_16X16X4_F32 per channel.
__global__ __launch_bounds__(256) void sh_eval_wmma_kernel(
    const float* __restrict__ coords,   // [N,3]
    const float* __restrict__ shc,      // [N*16, 2] : point-major, k-major, ch
    const float* __restrict__ rx,       // [3]
    float* __restrict__ out,            // [N,2]
    int n)
{
  const int lane = threadIdx.x & 31;
  const int col  = lane & 15;        // point within tile == A-matrix M == B/C N
  const bool h   = (lane & 16) != 0; // half-wave: which K elements this lane owns
  const int wid  = threadIdx.x >> 5;

  const int tile = blockIdx.x * (blockDim.x >> 5) + wid;
  const int base = tile * 16;
  if (base >= n) return;             // wave-uniform: EXEC stays all-ones below

  int q = base + col;
  const bool inrange = q < n;
  if (!inrange) q = n - 1;           // clamp (keep EXEC full for WMMA)

  // Kick off the 2KB coefficient tile early: per lane one 16B chunk per K-quad
  // (S[k][c] pairs for k, k+1 are contiguous), NT since the 256MB stream has
  // no reuse and exceeds the 192MB L2.
  const v4f* sp = (const v4f*)(shc + q * 32);
  const int hofs = h ? 1 : 0;
  const v4f s0 = __builtin_nontemporal_load(sp + 0 + hofs);
  const v4f s1 = __builtin_nontemporal_load(sp + 2 + hofs);
  const v4f s2 = __builtin_nontemporal_load(sp + 4 + hofs);
  const v4f s3 = __builtin_nontemporal_load(sp + 6 + hofs);

  const float* cp = coords + q * 3;
  const float dx0 = __builtin_nontemporal_load(cp + 0) - rx[0];
  const float dy0 = __builtin_nontemporal_load(cp + 1) - rx[1];
  const float dz0 = __builtin_nontemporal_load(cp + 2) - rx[2];
  const float inv = rsqrtf(dx0*dx0 + dy0*dy0 + dz0*dz0 + 1e-12f);
  const float x = dx0 * inv, y = dy0 * inv, z = dz0 * inv;

  const float xx = x*x, yy = y*y, zz = z*z;
  const float xy = x*y, yz = y*z, xz = x*z;

  // Named scalars (NOT an array) so the half-wave packing below stays a flat
  // one-bit select instead of a dynamic-index compare tree.
  const float b0  = 0.28209479177387814f;
  const float b1  = -0.4886025119029199f * y;
  const float b2  =  0.4886025119029199f * z;
  const float b3  = -0.4886025119029199f * x;
  const float b4  =  1.0925484305920792f * xy;
  const float b5  = -1.0925484305920792f * yz;
  const float b6  =  0.31539156525252005f * (2.0f*zz - xx - yy);
  const float b7  = -1.0925484305920792f * xz;
  const float b8  =  0.5462742152960396f * (xx - yy);
  const float b9  = -0.5900435899266435f * y * (3.0f*xx - yy);
  const float b10 =  2.890611442640554f  * xy * z;
  const float b11 = -0.4570457994644658f * y * (4.0f*zz - xx - yy);
  const float b12 =  0.3731763325901154f * z * (2.0f*zz - 3.0f*xx - 3.0f*yy);
  const float b13 = -0.4570457994644658f * x * (4.0f*zz - xx - yy);
  const float b14 =  1.445305721320277f  * z * (xx - yy);
  const float b15 = -0.5900435899266435f * x * (xx - 3.0f*yy);

  // A-matrix 16x4 f32 layout per step i: VGPR0 = K 4i+0 (lanes 0-15) / 4i+2
  // (lanes 16-31), VGPR1 = K 4i+1 / 4i+3.  One v_cndmask per component.
  v2f A0; A0.x = h ? b2  : b0;  A0.y = h ? b3  : b1;
  v2f A1; A1.x = h ? b6  : b4;  A1.y = h ? b7  : b5;
  v2f A2; A2.x = h ? b10 : b8;  A2.y = h ? b11 : b9;
  v2f A3; A3.x = h ? b14 : b12; A3.y = h ? b15 : b13;

  // B-matrix: VGPR0 = S[k], VGPR1 = S[k+1] for this lane's point (col) and
  // half's k; channel 0 uses .x/.z of the 16B chunk, channel 1 uses .y/.w.
  v2f B00; B00.x = s0.x; B00.y = s0.z;   v2f B01; B01.x = s0.y; B01.y = s0.w;
  v2f B10; B10.x = s1.x; B10.y = s1.z;   v2f B11; B11.x = s1.y; B11.y = s1.w;
  v2f B20; B20.x = s2.x; B20.y = s2.z;   v2f B21; B21.x = s2.y; B21.y = s2.w;
  v2f B30; B30.x = s3.x; B30.y = s3.z;   v2f B31; B31.x = s3.y; B31.y = s3.w;

  v8f acc0 = {0.f,0.f,0.f,0.f,0.f,0.f,0.f,0.f};
  v8f acc1 = {0.f,0.f,0.f,0.f,0.f,0.f,0.f,0.f};
  acc0 = __builtin_amdgcn_wmma_f32_16x16x4_f32(false, A0, false, B00, (short)0, acc0, false, false);
  acc1 = __builtin_amdgcn_wmma_f32_16x16x4_f32(false, A0, false, B01, (short)0, acc1, false, false);
  acc0 = __builtin_amdgcn_wmma_f32_16x16x4_f32(false, A1, false, B10, (short)0, acc0, false, false);
  acc1 = __builtin_amdgcn_wmma_f32_16x16x4_f32(false, A1, false, B11, (short)0, acc1, false, false);
  acc0 = __builtin_amdgcn_wmma_f32_16x16x4_f32(false, A2, false, B20, (short)0, acc0, false, false);
  acc1 = __builtin_amdgcn_wmma_f32_16x16x4_f32(false, A2, false, B21, (short)0, acc1, false, false);
  acc0 = __builtin_amdgcn_wmma_f32_16x16x4_f32(false, A3, false, B30, (short)0, acc0, false, false);
  acc1 = __builtin_amdgcn_wmma_f32_16x16x4_f32(false, A3, false, B31, (short)0, acc1, false, false);

  // Diagonal of D: point p lives in VGPR p&7 at lane p (p<8) / lane p+16.
  const int  r    = col & 7;
  const bool diag = ((lane >> 4) == (col >> 3));
  const float o0 = diag_pick8(acc0, r);
  const float o1 = diag_pick8(acc1, r);

  if (diag && inrange) {
    float* op = out + (base + col) * 2;
    __builtin_nontemporal_store(o0, op + 0);
    __builtin_nontemporal_store(o1, op + 1);
  }
}

extern "C" void kernel_launch(void* const* d_in, const int* in_sizes, int n_in,
                              void* d_out, int out_size, void* d_ws, size_t ws_size,
                              hipStream_t stream) {
  // setup_inputs order: coordinates, active_deg, max_coeffs, sh_coefficients, rx_pos
  const float* coords = (const float*)d_in[0];
  const float* shc    = (const float*)d_in[3];
  const float* rx     = (const float*)d_in[4];
  float* out = (float*)d_out;

  const int n = in_sizes[0] / 3;                 // 2,000,000
  const int waves_per_block = 256 / 32;
  const int tiles  = (n + 15) / 16;
  const int blocks = (tiles + waves_per_block - 1) / waves_per_block;

  sh_eval_wmma_kernel<<<blocks, 256, 0, stream>>>(coords, shc, rx, out, n);
}